// CRFTagModel_19619410608763
// MI455X (gfx1250) — compile-verified
//
#include <hip/hip_runtime.h>
#include <hip/hip_bf16.h>
#include <cstdint>

// Problem constants (reference: B,S,D,T = 64,512,768,64)
#define BB 64
#define SS 512
#define DD 768
#define TT 64

typedef float v2f __attribute__((ext_vector_type(2)));
typedef float v8f __attribute__((ext_vector_type(8)));

// ---------------------------------------------------------------------------
// Kernel 1: emissions GEMM  em[m][t] = sum_d x[m][d] * W[t][d] + bias[t]
//   M = B*S = 32768 rows, N = T = 64, K = D = 768
//   One wave per 16-row tile, computing all 4 column tiles (N=64).
//   Uses V_WMMA_F32_16X16X4_F32 (exact fp32 matrix op, CDNA5 wave32).
// ---------------------------------------------------------------------------
__global__ __launch_bounds__(256) void crf_emissions_wmma(
    const float* __restrict__ x,     // [M][DD]
    const float* __restrict__ W,     // [TT][DD]
    const float* __restrict__ bias,  // [TT]
    float* __restrict__ em)          // [M][TT]
{
  const int wave = blockIdx.x * (blockDim.x >> 5) + (threadIdx.x >> 5); // 0..2047
  const int lane = threadIdx.x & 31;
  const int m0   = wave * 16;             // row-tile base
  const int l16  = lane & 15;
  const int kh   = lane >> 4;             // 0 or 1 -> K offset 0 or 2

  // A: lane holds x[m0 + l16][k + 2*kh .. +1]   (ISA 32-bit A 16x4 layout)
  const float* __restrict__ arow = x + (size_t)(m0 + l16) * DD + 2 * kh;
  // B: lane holds W[ntile*16 + l16][k + 2*kh .. +1]  (B is W^T: B[k][n]=W[n][k])
  const float* __restrict__ w0 = W + (size_t)l16 * DD + 2 * kh;

  v8f acc0 = {0.f,0.f,0.f,0.f,0.f,0.f,0.f,0.f};
  v8f acc1 = acc0, acc2 = acc0, acc3 = acc0;

  #pragma unroll 4
  for (int k = 0; k < DD; k += 4) {
    v2f a  = *(const v2f*)(arow + k);
    v2f b0 = *(const v2f*)(w0 + k);
    v2f b1 = *(const v2f*)(w0 + 16 * DD + k);
    v2f b2 = *(const v2f*)(w0 + 32 * DD + k);
    v2f b3 = *(const v2f*)(w0 + 48 * DD + k);
    acc0 = __builtin_amdgcn_wmma_f32_16x16x4_f32(false, a, false, b0, (short)0, acc0, false, false);
    acc1 = __builtin_amdgcn_wmma_f32_16x16x4_f32(false, a, false, b1, (short)0, acc1, false, false);
    acc2 = __builtin_amdgcn_wmma_f32_16x16x4_f32(false, a, false, b2, (short)0, acc2, false, false);
    acc3 = __builtin_amdgcn_wmma_f32_16x16x4_f32(false, a, false, b3, (short)0, acc3, false, false);
  }

  // C/D layout: VGPR v -> row m0 + v + (lane>=16 ? 8 : 0), col ntile*16 + (lane&15)
  const int rbase = m0 + kh * 8;
  #pragma unroll
  for (int nt = 0; nt < 4; ++nt) {
    v8f acc = (nt == 0) ? acc0 : (nt == 1) ? acc1 : (nt == 2) ? acc2 : acc3;
    const int n  = nt * 16 + l16;
    const float bn = bias[n];
    #pragma unroll
    for (int v = 0; v < 8; ++v) {
      em[(size_t)(rbase + v) * TT + n] = acc[v] + bn;
    }
  }
}

// ---------------------------------------------------------------------------
// Kernel 2: Viterbi forward scan + backtrack, one workgroup per batch element.
//   256 threads: tc = tid & 63 (current tag), p = tid >> 6 (prev-tag slice).
//   Transition column slice kept in registers; scores + backpointer history
//   live entirely in LDS (511*64 u8 = 32 KB of the 320 KB WGP LDS).
//   Tie-breaking matches jnp.argmax (first occurrence): strict '>' ascending.
// ---------------------------------------------------------------------------
__global__ __launch_bounds__(256) void crf_viterbi(
    const float* __restrict__ em,     // [BB][SS][TT]
    const float* __restrict__ trans,  // [TT][TT]
    int* __restrict__ out)            // [BB][SS]
{
  const int b   = blockIdx.x;
  const int tid = threadIdx.x;
  const int tc  = tid & 63;
  const int p   = tid >> 6;          // 0..3, prev-tag slice [p*16, p*16+16)

  __shared__ float         score[TT];
  __shared__ float         pval[4][TT];
  __shared__ unsigned char pidx[4][TT];
  __shared__ unsigned char hist[(SS - 1) * TT];   // backpointers

  // Register-resident transition slice: tr[i] = trans[p*16+i][tc]
  float tr[16];
  #pragma unroll
  for (int i = 0; i < 16; ++i) tr[i] = trans[(p * 16 + i) * TT + tc];

  const float* __restrict__ emb = em + (size_t)b * SS * TT;

  if (tid < TT) score[tc] = emb[tc];   // score_0 = em[t=0]
  __syncthreads();

  for (int t = 1; t < SS; ++t) {
    // Partial max over this thread's 16 prev tags
    float best = -__builtin_inff();
    int   bi   = p * 16;
    #pragma unroll
    for (int i = 0; i < 16; ++i) {
      float v = score[p * 16 + i] + tr[i];
      if (v > best) { best = v; bi = p * 16 + i; }
    }
    pval[p][tc] = best;
    pidx[p][tc] = (unsigned char)bi;
    __syncthreads();

    if (p == 0) {
      float bv = pval[0][tc];
      int   bb = pidx[0][tc];
      #pragma unroll
      for (int q = 1; q < 4; ++q) {
        float v = pval[q][tc];
        if (v > bv) { bv = v; bb = pidx[q][tc]; }   // earlier slice wins ties
      }
      score[tc] = bv + emb[(size_t)t * TT + tc];
      hist[(t - 1) * TT + tc] = (unsigned char)bb;
    }
    __syncthreads();
  }

  // Final argmax + backtrack (sequential, one thread; history is in LDS)
  if (tid == 0) {
    float bv = score[0];
    int   bt = 0;
    for (int t2 = 1; t2 < TT; ++t2) {
      if (score[t2] > bv) { bv = score[t2]; bt = t2; }
    }
    int* __restrict__ ob = out + (size_t)b * SS;
    ob[SS - 1] = bt;
    int cur = bt;
    for (int t = SS - 2; t >= 0; --t) {
      cur = hist[t * TT + cur];
      ob[t] = cur;
    }
  }
}

// ---------------------------------------------------------------------------
// Launch: inputs in setup_inputs() order: x [B,S,D] f32, W [T,D] f32,
// b [T] f32, transitions [T,T] f32. Output: int32 [B,S].
// Workspace: em [B*S*T] f32 = 8.4 MB.
// ---------------------------------------------------------------------------
extern "C" void kernel_launch(void* const* d_in, const int* in_sizes, int n_in,
                              void* d_out, int out_size, void* d_ws, size_t ws_size,
                              hipStream_t stream) {
  const float* x     = (const float*)d_in[0];
  const float* W     = (const float*)d_in[1];
  const float* bias  = (const float*)d_in[2];
  const float* trans = (const float*)d_in[3];
  float* em = (float*)d_ws;              // [B*S][T] = 32768 * 64 floats
  int*   out = (int*)d_out;              // [B][S] int32

  // GEMM: 2048 row-tiles, 8 waves (256 threads) per block -> 256 blocks
  crf_emissions_wmma<<<(BB * SS / 16) / 8, 256, 0, stream>>>(x, W, bias, em);
  // Viterbi: one workgroup per batch element
  crf_viterbi<<<BB, 256, 0, stream>>>(em, trans, out);
}